// MoeMuxExpertChoiceKTokens_33552284516484
// MI455X (gfx1250) — compile-verified
//
#include <hip/hip_runtime.h>
#include <cstdint>
#include <cstddef>

// Problem constants (from reference)
#define BB 64
#define SS 2048
#define DD 1024
#define EE 32
#define HH 4096
#define KK 8

typedef __attribute__((ext_vector_type(2))) float v2f;
typedef __attribute__((ext_vector_type(8))) float v8f;

// D = A(16x4) * B(4x16) + C(16x16), fp32 WMMA.
// Layout assumption (per CDNA5 ISA):
//  A: lane<16 -> row=lane, v0=K0,v1=K1 ; lane>=16 -> row=lane-16, v0=K2,v1=K3
//  B: lane<16 -> col=lane, v0=K0,v1=K1 ; lane>=16 -> col=lane-16, v0=K2,v1=K3
//  C: vgpr r -> row = r + 8*(lane>=16), col = lane&15
static __device__ __forceinline__ v8f wmma_f32(v2f a, v2f b, v8f c) {
  return __builtin_amdgcn_wmma_f32_16x16x4_f32(
      /*neg_a=*/false, a, /*neg_b=*/false, b,
      /*c_mod=*/(short)0, c, /*reuse_a=*/false, /*reuse_b=*/false);
}

// ---------------------------------------------------------------------------
// Stage 1: gating logits. One wave per 16-token tile, all 32 experts.
// Writes logits transposed: logits_t[b][e][s]  (coalesced softmax reads).
// ---------------------------------------------------------------------------
__global__ __launch_bounds__(32) void gate_gemm(
    const float* __restrict__ x, const float* __restrict__ gw,
    const float* __restrict__ gb, float* __restrict__ logits_t) {
  const int tile = blockIdx.x;            // 16 tokens per tile
  const int lane = threadIdx.x;
  const int half = lane >> 4;             // K sub-phase (0: K0/K1, 1: K2/K3)
  const int l15  = lane & 15;

  const float* arow = x + ((size_t)tile * 16 + l15) * DD + 2 * half;
  const float* b0   = gw + (size_t)l15 * DD + 2 * half;         // experts 0..15
  const float* b1   = gw + (size_t)(l15 + 16) * DD + 2 * half;  // experts 16..31

  v8f acc0 = {};
  v8f acc1 = {};
  for (int k = 0; k < DD; k += 4) {
    v2f a  = *(const v2f*)(arow + k);
    v2f w0 = *(const v2f*)(b0 + k);
    v2f w1v = *(const v2f*)(b1 + k);
    acc0 = wmma_f32(a, w0, acc0);
    acc1 = wmma_f32(a, w1v, acc1);
  }
  const float bias0 = gb[l15];
  const float bias1 = gb[l15 + 16];
  const int rowbase = tile * 16 + half * 8;
#pragma unroll
  for (int r = 0; r < 8; ++r) {
    const int t  = rowbase + r;       // global token index over B*S
    const int bI = t >> 11;           // /S
    const int s  = t & (SS - 1);
    logits_t[((size_t)bI * EE + l15) * SS + s]        = acc0[r] + bias0;
    logits_t[((size_t)bI * EE + (l15 + 16)) * SS + s] = acc1[r] + bias1;
  }
}

// ---------------------------------------------------------------------------
// Stage 2: softmax over tokens + top-K per (b,e). One wave per (b,e).
// ---------------------------------------------------------------------------
__global__ __launch_bounds__(256) void softmax_topk(
    const float* __restrict__ logits_t, float* __restrict__ vals,
    int* __restrict__ idxb) {
  const int wave = (int)((blockIdx.x * blockDim.x + threadIdx.x) >> 5);
  const int lane = threadIdx.x & 31;
  const int b = wave / EE;
  const int e = wave % EE;
  const float* row = logits_t + (size_t)wave * SS;   // wave == b*EE+e

  float v[SS / 32];
  float mx = -3.4e38f;
#pragma unroll
  for (int i = 0; i < SS / 32; ++i) {
    v[i] = row[i * 32 + lane];
    mx = fmaxf(mx, v[i]);
  }
#pragma unroll
  for (int off = 16; off; off >>= 1) mx = fmaxf(mx, __shfl_xor(mx, off, 32));

  float sum = 0.0f;
#pragma unroll
  for (int i = 0; i < SS / 32; ++i) sum += expf(v[i] - mx);
#pragma unroll
  for (int off = 16; off; off >>= 1) sum += __shfl_xor(sum, off, 32);
  const float rinv = 1.0f / sum;

  int chosen[KK];
#pragma unroll
  for (int j = 0; j < KK; ++j) chosen[j] = -1;

#pragma unroll
  for (int r = 0; r < KK; ++r) {
    float lv = -3.4e38f;
    int li = SS;
#pragma unroll
    for (int i = 0; i < SS / 32; ++i) {
      const int gi = i * 32 + lane;
      bool skip = false;
#pragma unroll
      for (int j = 0; j < KK; ++j) skip = skip || (chosen[j] == gi);
      if (!skip && v[i] > lv) { lv = v[i]; li = gi; }
    }
#pragma unroll
    for (int off = 16; off; off >>= 1) {
      const float ov = __shfl_xor(lv, off, 32);
      const int   oi = __shfl_xor(li, off, 32);
      if (ov > lv || (ov == lv && oi < li)) { lv = ov; li = oi; }
    }
    chosen[r] = li;
    if (lane == 0) {
      vals[((size_t)b * KK + r) * EE + e] = expf(lv - mx) * rinv;
      idxb[((size_t)b * KK + r) * EE + e] = li;
    }
  }
}

// ---------------------------------------------------------------------------
// Stage 3: weighted merge of K gathered tokens -> merged[E][B][D].
// ---------------------------------------------------------------------------
__global__ __launch_bounds__(256) void merge_tokens(
    const float* __restrict__ x, const float* __restrict__ vals,
    const int* __restrict__ idxb, float* __restrict__ merged) {
  const int b = blockIdx.x / EE;
  const int e = blockIdx.x % EE;
  const int d0 = threadIdx.x * 4;
  float a0 = 0.f, a1 = 0.f, a2 = 0.f, a3 = 0.f;
#pragma unroll
  for (int k = 0; k < KK; ++k) {
    const float w = vals[((size_t)b * KK + k) * EE + e];
    const int   t = idxb[((size_t)b * KK + k) * EE + e];
    const float4 xv = *(const float4*)(x + ((size_t)b * SS + t) * DD + d0);
    a0 += w * xv.x; a1 += w * xv.y; a2 += w * xv.z; a3 += w * xv.w;
  }
  float4 o; o.x = a0; o.y = a1; o.z = a2; o.w = a3;
  *(float4*)(merged + ((size_t)e * BB + b) * DD + d0) = o;
}

// ---------------------------------------------------------------------------
// Stage 4: FFN layer 1 + exact GELU.  h[e][64][H] = gelu([merged,1] @ w1[e]).
// One wave per (e, 16-col group): 4 M-tiles x 1 N-tile, w1 streamed once.
// ---------------------------------------------------------------------------
__global__ __launch_bounds__(32) void ffn1(
    const float* __restrict__ merged, const float* __restrict__ w1,
    float* __restrict__ h) {
  const int NG = HH / 16;   // 256
  const int e  = blockIdx.x / NG;
  const int ng = blockIdx.x % NG;
  const int lane = threadIdx.x;
  const int half = lane >> 4;
  const int l15  = lane & 15;
  const int n    = ng * 16 + l15;

  const float* A  = merged + (size_t)e * BB * DD + 2 * half;
  const float* Wb = w1 + (size_t)e * (DD + 1) * HH + n;

  v8f acc[4] = {};
  for (int k = 0; k < DD; k += 4) {
    const int kk = k + 2 * half;
    v2f a0 = *(const v2f*)(A + (size_t)(l15)      * DD + k);
    v2f a1 = *(const v2f*)(A + (size_t)(16 + l15) * DD + k);
    v2f a2 = *(const v2f*)(A + (size_t)(32 + l15) * DD + k);
    v2f a3 = *(const v2f*)(A + (size_t)(48 + l15) * DD + k);
    v2f bv;
    bv.x = Wb[(size_t)kk * HH];
    bv.y = Wb[(size_t)(kk + 1) * HH];
    acc[0] = wmma_f32(a0, bv, acc[0]);
    acc[1] = wmma_f32(a1, bv, acc[1]);
    acc[2] = wmma_f32(a2, bv, acc[2]);
    acc[3] = wmma_f32(a3, bv, acc[3]);
  }
  { // tail: K = 1024 is the ones column / bias row of w1; 1025..1027 are zero
    v2f a; a.x = (half == 0) ? 1.0f : 0.0f; a.y = 0.0f;
    v2f bv; bv.x = (half == 0) ? Wb[(size_t)DD * HH] : 0.0f; bv.y = 0.0f;
#pragma unroll
    for (int mt = 0; mt < 4; ++mt) acc[mt] = wmma_f32(a, bv, acc[mt]);
  }
#pragma unroll
  for (int mt = 0; mt < 4; ++mt) {
#pragma unroll
    for (int r = 0; r < 8; ++r) {
      const int row = mt * 16 + half * 8 + r;
      const float xv = acc[mt][r];
      const float g = 0.5f * xv * (1.0f + erff(xv * 0.7071067811865476f));
      h[((size_t)e * BB + row) * HH + n] = g;
    }
  }
}

// ---------------------------------------------------------------------------
// Stage 5: FFN layer 2.  o[e][64][D] = [h,1] @ w2[e].  w2 streamed once.
// ---------------------------------------------------------------------------
__global__ __launch_bounds__(32) void ffn2(
    const float* __restrict__ h, const float* __restrict__ w2,
    float* __restrict__ o) {
  const int NG = DD / 16;   // 64
  const int e  = blockIdx.x / NG;
  const int ng = blockIdx.x % NG;
  const int lane = threadIdx.x;
  const int half = lane >> 4;
  const int l15  = lane & 15;
  const int n    = ng * 16 + l15;

  const float* A  = h + (size_t)e * BB * HH + 2 * half;
  const float* Wb = w2 + (size_t)e * (HH + 1) * DD + n;

  v8f acc[4] = {};
  for (int k = 0; k < HH; k += 4) {
    const int kk = k + 2 * half;
    v2f a0 = *(const v2f*)(A + (size_t)(l15)      * HH + k);
    v2f a1 = *(const v2f*)(A + (size_t)(16 + l15) * HH + k);
    v2f a2 = *(const v2f*)(A + (size_t)(32 + l15) * HH + k);
    v2f a3 = *(const v2f*)(A + (size_t)(48 + l15) * HH + k);
    v2f bv;
    bv.x = Wb[(size_t)kk * DD];
    bv.y = Wb[(size_t)(kk + 1) * DD];
    acc[0] = wmma_f32(a0, bv, acc[0]);
    acc[1] = wmma_f32(a1, bv, acc[1]);
    acc[2] = wmma_f32(a2, bv, acc[2]);
    acc[3] = wmma_f32(a3, bv, acc[3]);
  }
  { // tail: K = 4096 ones column / bias row of w2
    v2f a; a.x = (half == 0) ? 1.0f : 0.0f; a.y = 0.0f;
    v2f bv; bv.x = (half == 0) ? Wb[(size_t)HH * DD] : 0.0f; bv.y = 0.0f;
#pragma unroll
    for (int mt = 0; mt < 4; ++mt) acc[mt] = wmma_f32(a, bv, acc[mt]);
  }
#pragma unroll
  for (int mt = 0; mt < 4; ++mt) {
#pragma unroll
    for (int r = 0; r < 8; ++r) {
      const int row = mt * 16 + half * 8 + r;
      o[((size_t)e * BB + row) * DD + n] = acc[mt][r];
    }
  }
}

// ---------------------------------------------------------------------------
// Stage 6: demux scatter-add. out[b, idx[b,k,e], :] += vals[b,k,e]*o[e][b][:]
// ---------------------------------------------------------------------------
__global__ __launch_bounds__(256) void scatter_out(
    const float* __restrict__ o, const float* __restrict__ vals,
    const int* __restrict__ idxb, float* __restrict__ out) {
  const int b = blockIdx.x / EE;
  const int e = blockIdx.x % EE;
  const int d0 = threadIdx.x * 4;
  const float4 ov = *(const float4*)(o + ((size_t)e * BB + b) * DD + d0);
#pragma unroll
  for (int k = 0; k < KK; ++k) {
    const float w = vals[((size_t)b * KK + k) * EE + e];
    const int   t = idxb[((size_t)b * KK + k) * EE + e];
    float* dst = out + ((size_t)b * SS + t) * DD + d0;
    __hip_atomic_fetch_add(dst + 0, w * ov.x, __ATOMIC_RELAXED, __HIP_MEMORY_SCOPE_AGENT);
    __hip_atomic_fetch_add(dst + 1, w * ov.y, __ATOMIC_RELAXED, __HIP_MEMORY_SCOPE_AGENT);
    __hip_atomic_fetch_add(dst + 2, w * ov.z, __ATOMIC_RELAXED, __HIP_MEMORY_SCOPE_AGENT);
    __hip_atomic_fetch_add(dst + 3, w * ov.w, __ATOMIC_RELAXED, __HIP_MEMORY_SCOPE_AGENT);
  }
}

// ---------------------------------------------------------------------------
extern "C" void kernel_launch(void* const* d_in, const int* in_sizes, int n_in,
                              void* d_out, int out_size, void* d_ws,
                              size_t ws_size, hipStream_t stream) {
  (void)in_sizes; (void)n_in; (void)out_size; (void)ws_size;
  const float* x  = (const float*)d_in[0];
  const float* gw = (const float*)d_in[1];
  const float* gb = (const float*)d_in[2];
  const float* w1 = (const float*)d_in[3];
  const float* w2 = (const float*)d_in[4];
  float* out = (float*)d_out;

  // Workspace layout (~64.5 MB)
  char* ws = (char*)d_ws;
  float* logits_t = (float*)ws;                                   // B*E*S f32 (16 MB)
  size_t off = (size_t)BB * EE * SS * sizeof(float);
  float* vals = (float*)(ws + off);  off += (size_t)BB * KK * EE * sizeof(float);
  int*   idxb = (int*)(ws + off);    off += (size_t)BB * KK * EE * sizeof(int);
  float* merged = (float*)(ws + off); off += (size_t)EE * BB * DD * sizeof(float);
  float* h = (float*)(ws + off);      off += (size_t)EE * BB * HH * sizeof(float);
  float* o = (float*)(ws + off);

  hipMemsetAsync(out, 0, (size_t)BB * SS * DD * sizeof(float), stream);

  gate_gemm<<<(BB * SS) / 16, 32, 0, stream>>>(x, gw, gb, logits_t);
  softmax_topk<<<(BB * EE) / 8, 256, 0, stream>>>(logits_t, vals, idxb);
  merge_tokens<<<BB * EE, 256, 0, stream>>>(x, vals, idxb, merged);
  ffn1<<<EE * (HH / 16), 32, 0, stream>>>(merged, w1, h);
  ffn2<<<EE * (DD / 16), 32, 0, stream>>>(h, w2, o);
  scatter_out<<<BB * EE, 256, 0, stream>>>(o, vals, idxb, out);
}